// LinearStateSpaceLayer_1958505087206
// MI455X (gfx1250) — compile-verified
//
#include <hip/hip_runtime.h>

// ---------------------------------------------------------------------------
// Diagonal LTI SSM scan, chunked into 16-step blocks mapped onto CDNA5
// V_WMMA_F32_16X16X4_F32 tiles (time = M dim, batch = N dim, state/time = K).
// One wave per hidden channel d; 16 channels per 512-thread workgroup so all
// global traffic is coalesced 16B along the contiguous d axis.
// Per chunk (branch-free inner loop):  Y = M*U + P*X ;  X = Lam.*X + G*U.
// ---------------------------------------------------------------------------

typedef __attribute__((ext_vector_type(2))) float v2f;
typedef __attribute__((ext_vector_type(4))) float v4f;
typedef __attribute__((ext_vector_type(8))) float v8f;

#define BATCH  8
#define SEQLEN 4096
#define HID    1024
#define STATE  64
#define T      16
#define NCHUNK (SEQLEN / T)   // 256
#define DPB    16             // d-channels per block (one wave each)
#define BP     16             // batch columns padded to WMMA width

// Precomputed per-channel parameter pack (floats)
#define PER_D   2368
#define OFF_P   0             // P[t][n]   16x64 : C * dA^(t+1)
#define OFF_G   1024          // G[n][s]   64x16 : dB * dA^(15-s)
#define OFF_M   2048          // M[t][s]   16x16 : causal Toeplitz of Kloc (+Dv diag)
#define OFF_LAM 2304          // Lam[n]    64    : dA^16

static __device__ __forceinline__ v8f wmma4(v2f a, v2f b, v8f c) {
  // D = A(16x4,f32) * B(4x16,f32) + C(16x16,f32)
  return __builtin_amdgcn_wmma_f32_16x16x4_f32(false, a, false, b, (short)0, c,
                                               false, false);
}

// ---------------------------------------------------------------------------
// Setup: derive ZOH-discretized per-channel matrices. 1024 blocks x 64 threads.
// ---------------------------------------------------------------------------
__global__ __launch_bounds__(64) void ssm_setup_kernel(
    const float* __restrict__ A, const float* __restrict__ Bm,
    const float* __restrict__ C, const float* __restrict__ Dv,
    const float* __restrict__ dt, float* __restrict__ ws) {
  const int d = blockIdx.x;
  const int n = threadIdx.x;  // state index 0..63
  __shared__ float sK[T * STATE];
  __shared__ float sKl[T];
  float* base = ws + (size_t)d * PER_D;

  const float a  = expf(A[d * STATE + n]);          // -A_cont (positive)
  const float da = expf(-a * dt[n]);                // dA
  const float db = Bm[d * STATE + n] * (1.0f - da) / a;
  const float cc = C[d * STATE + n];

  // P[t][n] = C * dA^(t+1)
  float p = da;
#pragma unroll
  for (int t = 0; t < T; ++t) { base[OFF_P + t * STATE + n] = cc * p; p *= da; }

  // Lam[n] = dA^16
  const float da2 = da * da, da4 = da2 * da2, da8 = da4 * da4;
  base[OFF_LAM + n] = da8 * da8;

  // G[n][s] = dB * dA^(15-s)
  float g = db;
  for (int s = T - 1; s >= 0; --s) { base[OFF_G + n * T + s] = g; g *= da; }

  // Kloc[tau] = sum_n C*dB*dA^tau  (block reduction over n)
  float w = cc * db;
#pragma unroll
  for (int tau = 0; tau < T; ++tau) { sK[tau * STATE + n] = w; w *= da; }
  __syncthreads();
  if (n < T) {
    float s = 0.0f;
    for (int m = 0; m < STATE; ++m) s += sK[n * STATE + m];
    sKl[n] = s;
  }
  __syncthreads();
  if (n < T) {
    const int t = n;
    const float dvd = Dv[d];
    for (int s2 = 0; s2 < T; ++s2) {
      float v = (t >= s2) ? sKl[t - s2] : 0.0f;
      if (s2 == t) v += dvd;  // feed-through on the diagonal
      base[OFF_M + t * T + s2] = v;
    }
  }
}

// ---------------------------------------------------------------------------
// Main scan: 64 blocks x 512 threads (16 waves). Wave wv owns channel
// d = blockIdx.x*16 + wv; loops over 256 chunks keeping state X in WMMA
// C/D-layout registers.
// ---------------------------------------------------------------------------
__global__ __launch_bounds__(32 * DPB) void ssm_scan_kernel(
    const float* __restrict__ u, const float* __restrict__ ws,
    float* __restrict__ out) {
  const int tid  = threadIdx.x;
  const int lane = tid & 31;
  const int wv   = tid >> 5;        // d_local = wave id
  const int d0   = blockIdx.x * DPB;
  const int half = lane >> 4;       // WMMA lane half
  const int col  = lane & 15;       // batch column (8 used, 8 zero-padded)

  // Batch-padded tiles: [t][b16][d16]; columns 8..15 of ldsU are zeroed once
  // so all inner-loop DS accesses are unconditional (no EXEC branching).
  __shared__ __align__(16) float ldsU[T * BP * DPB];  // 16KB
  __shared__ __align__(16) float ldsY[T * BP * DPB];  // 16KB

  // zero the padding (and everything else) once
#pragma unroll
  for (int k = 0; k < (T * BP * DPB) / (32 * DPB); ++k)
    ldsU[tid + k * (32 * DPB)] = 0.0f;

  const float* base = ws + (size_t)(d0 + wv) * PER_D;

  // ---- A-operands, register-resident for all 256 chunks -------------------
  // f32 16x4 A layout: lane (m=col, k-half=half), vgpr j holds K = 2*half+j.
  v2f pA[16];
#pragma unroll
  for (int kk = 0; kk < 16; ++kk)
    pA[kk] = *(const v2f*)(base + OFF_P + col * STATE + 4 * kk + 2 * half);
  v2f mA[4];
#pragma unroll
  for (int kk = 0; kk < 4; ++kk)
    mA[kk] = *(const v2f*)(base + OFF_M + col * T + 4 * kk + 2 * half);
  v2f gA[4][4];
#pragma unroll
  for (int r = 0; r < 4; ++r)
#pragma unroll
    for (int kk = 0; kk < 4; ++kk)
      gA[r][kk] =
          *(const v2f*)(base + OFF_G + (r * 16 + col) * T + 4 * kk + 2 * half);
  // Lam in C/D layout: tile r, vgpr i -> n = r*16 + i + 8*half
  float lam[4][8];
#pragma unroll
  for (int r = 0; r < 4; ++r)
#pragma unroll
    for (int i = 0; i < 8; ++i)
      lam[r][i] = base[OFF_LAM + r * 16 + i + 8 * half];

  v8f x[4] = {};  // carried state, C/D layout (rows n, cols b)

  // cooperative tile mapping: 512 threads <-> 8b x 16t x (4 float4 along d)
  const int pair = tid >> 2;        // b*16 + t
  const int bb   = pair >> 4;       // 0..7
  const int tt   = pair & 15;
  const int d4   = (tid & 3) << 2;
  const size_t gRow = (size_t)bb * SEQLEN * HID + (size_t)(d0 + d4);

  for (int c = 0; c < NCHUNK; ++c) {
    {  // stage u chunk tile, coalesced 16B along d; prefetch next chunk
      const float* gp = u + gRow + (size_t)(c * T + tt) * HID;
      v4f v = *(const v4f*)gp;
      *(v4f*)(&ldsU[(tt * BP + bb) * DPB + d4]) = v;
      if (c + 1 < NCHUNK)
        __builtin_prefetch(u + gRow + (size_t)((c + 1) * T + tt) * HID, 0, 0);
    }
    __syncthreads();  // also orders prev chunk's ldsY readback vs new writes

    // ---- U (rows = time s, cols = batch) as f32 4x16 B-operand blocks
    v2f uB[4];
#pragma unroll
    for (int kk = 0; kk < 4; ++kk) {
#pragma unroll
      for (int j = 0; j < 2; ++j) {
        const int s = 4 * kk + 2 * half + j;
        uB[kk][j] = ldsU[(s * BP + col) * DPB + wv];  // unconditional
      }
    }

    // ---- Y = M * U  (intra-chunk causal convolution)
    v8f y = {};
#pragma unroll
    for (int kk = 0; kk < 4; ++kk) y = wmma4(mA[kk], uB[kk], y);

    // ---- Y += P * X (carried state); C/D-layout X -> B-operand via shuffles
#pragma unroll
    for (int kk = 0; kk < 16; ++kk) {
      v2f xb;
#pragma unroll
      for (int j = 0; j < 2; ++j) {
        const int n0 = 4 * kk + j;       // value needed by lower lane-half
        const int n1 = 4 * kk + 2 + j;   // value needed by upper lane-half
        const float c0 =
            __shfl(x[n0 >> 4][(n0 & 15) & 7], col + 16 * ((n0 & 15) >> 3), 32);
        const float c1 =
            __shfl(x[n1 >> 4][(n1 & 15) & 7], col + 16 * ((n1 & 15) >> 3), 32);
        xb[j] = half ? c1 : c0;
      }
      y = wmma4(pA[kk], xb, y);
    }

    // ---- X = Lam .* X + G * U  (chunk-exit state)
#pragma unroll
    for (int r = 0; r < 4; ++r) {
      v8f xn;
#pragma unroll
      for (int i = 0; i < 8; ++i) xn[i] = lam[r][i] * x[r][i];
#pragma unroll
      for (int kk = 0; kk < 4; ++kk) xn = wmma4(gA[r][kk], uB[kk], xn);
      x[r] = xn;
    }

    // ---- stage Y to LDS (unconditional, padded cols), coalesced store
#pragma unroll
    for (int i = 0; i < 8; ++i) {
      const int t = i + 8 * half;
      ldsY[(t * BP + col) * DPB + wv] = y[i];
    }
    __syncthreads();
    {
      v4f v = *(const v4f*)(&ldsY[(tt * BP + bb) * DPB + d4]);
      *(v4f*)(out + gRow + (size_t)(c * T + tt) * HID) = v;
    }
  }
}

// ---------------------------------------------------------------------------
extern "C" void kernel_launch(void* const* d_in, const int* in_sizes, int n_in,
                              void* d_out, int out_size, void* d_ws,
                              size_t ws_size, hipStream_t stream) {
  (void)in_sizes; (void)n_in; (void)out_size; (void)ws_size;
  const float* u  = (const float*)d_in[0];
  const float* A  = (const float*)d_in[1];
  const float* Bm = (const float*)d_in[2];
  const float* C  = (const float*)d_in[3];
  const float* Dv = (const float*)d_in[4];
  const float* dt = (const float*)d_in[5];
  float* out = (float*)d_out;
  float* ws  = (float*)d_ws;  // needs HID*PER_D*4 ~= 9.3 MB scratch

  ssm_setup_kernel<<<HID, STATE, 0, stream>>>(A, Bm, C, Dv, dt, ws);
  ssm_scan_kernel<<<HID / DPB, 32 * DPB, 0, stream>>>(u, ws, out);
}